// GCN_63333587747173
// MI455X (gfx1250) — compile-verified
//
#include <hip/hip_runtime.h>

// GCN forward for MI455X (gfx1250, wave32).
// Strategy: fold symmetric normalization into GEMM epilogue, use
// V_WMMA_F32_16X16X4_F32 for all dense GEMMs (exact fp32), wave-per-edge
// coalesced scatter with hardware f32 atomics (resolve in the 192MB L2;
// whole working set ~52MB is L2-resident).

#define NN 50000     // N_NODES  (divisible by 16: 3125 tiles)
#define NE 800000    // N_EDGES
#define NG 512       // NUM_GRAPHS

typedef float v2f __attribute__((ext_vector_type(2)));
typedef float v8f __attribute__((ext_vector_type(8)));

__device__ __forceinline__ void atomAddF(float* p, float v) {
  unsafeAtomicAdd(p, v);  // -> global_atomic_add_f32 (no CAS loop)
}

// ---------------------------------------------------------------------------
// One 16x16 f32 tile of A(16xK, row-major, stride K) x B(Kx16 slice of KxNC).
// Fragment layouts per CDNA5 ISA 7.12.2 (32-bit A 16x4 / B / C-D tables):
//   A: lane&15 = M; VGPR0 holds K = k + 2*(lane>>4), VGPR1 = +1
//   B: lane&15 = N; VGPR0 holds K = k + 2*(lane>>4), VGPR1 = +1
//   D: VGPR j -> row (j + 8*(lane>>4)), col = lane&15
// ---------------------------------------------------------------------------
template <int K>
__device__ __forceinline__ v8f wmma_tile_f32(const float* __restrict__ A,
                                             const float* __restrict__ B,
                                             int ldb) {
  const int lane = threadIdx.x & 31;
  const int mn = lane & 15;
  const int kh = (lane >> 4) << 1;
  v8f acc = {};
#pragma unroll
  for (int k = 0; k < K; k += 4) {
    v2f a, b;
    a.x = A[mn * K + k + kh];
    a.y = A[mn * K + k + kh + 1];
    b.x = B[(k + kh) * ldb + mn];
    b.y = B[(k + kh + 1) * ldb + mn];
    acc = __builtin_amdgcn_wmma_f32_16x16x4_f32(false, a, false, b, (short)0,
                                                acc, false, false);
  }
  return acc;
}

// hn = (X @ W) * dinv[row]; also agg = hn (self-loop term pre-seeded).
template <int K, int NC>
__global__ void gemm_scaled_dual(const float* __restrict__ X,
                                 const float* __restrict__ W,
                                 const float* __restrict__ dinv,
                                 float* __restrict__ hn,
                                 float* __restrict__ agg) {
  const int mBase = blockIdx.x * 16;
  const int nBase = blockIdx.y * 16;
  v8f acc = wmma_tile_f32<K>(X + (size_t)mBase * K, W + nBase, NC);
  const int lane = threadIdx.x & 31;
  const int n = lane & 15;
  const int rOff = (lane >> 4) << 3;
#pragma unroll
  for (int j = 0; j < 8; ++j) {
    const int row = mBase + j + rOff;
    const float v = acc[j] * dinv[row];
    const size_t idx = (size_t)row * NC + nBase + n;
    hn[idx] = v;
    agg[idx] = v;
  }
}

// out = A @ B + bias  (final classifier head)
template <int K, int NC>
__global__ void gemm_bias(const float* __restrict__ A,
                          const float* __restrict__ B,
                          const float* __restrict__ bias,
                          float* __restrict__ out) {
  const int mBase = blockIdx.x * 16;
  const int nBase = blockIdx.y * 16;
  v8f acc = wmma_tile_f32<K>(A + (size_t)mBase * K, B + nBase, NC);
  const int lane = threadIdx.x & 31;
  const int n = lane & 15;
  const int rOff = (lane >> 4) << 3;
#pragma unroll
  for (int j = 0; j < 8; ++j) {
    const int row = mBase + j + rOff;
    out[(size_t)row * NC + nBase + n] = acc[j] + bias[nBase + n];
  }
}

// ---------------------------------------------------------------------------
// Edge scatter: one wave per edge; lane covers VEC consecutive features.
// Wave gather of hn[src][*] is a fully-coalesced 32*VEC*4-byte read.
// ---------------------------------------------------------------------------
template <int VEC>
__global__ void scatter_edges(const int* __restrict__ srcI,
                              const int* __restrict__ dstI,
                              const float* __restrict__ hn,
                              float* __restrict__ agg) {
  const int tid = blockIdx.x * blockDim.x + threadIdx.x;
  const int e = tid >> 5;
  const int lane = tid & 31;
  if (e >= NE) return;
  constexpr int F = 32 * VEC;
  const int s = srcI[e];
  const int d = dstI[e];
  const float* hs = hn + (size_t)s * F + lane * VEC;
  float* ad = agg + (size_t)d * F + lane * VEC;
#pragma unroll
  for (int i = 0; i < VEC; ++i) atomAddF(ad + i, hs[i]);
}

// h = [relu](agg * dinv[row] + bias[col])
template <int F, bool RELU>
__global__ void finalize_nodes(const float* __restrict__ agg,
                               const float* __restrict__ dinv,
                               const float* __restrict__ bias,
                               float* __restrict__ out) {
  const int i = blockIdx.x * blockDim.x + threadIdx.x;
  if (i >= NN * F) return;
  const int row = i / F;       // F is power-of-two -> shift
  const int col = i & (F - 1);
  float v = agg[i] * dinv[row] + bias[col];
  if (RELU) v = fmaxf(v, 0.0f);
  out[i] = v;
}

// ------------------------- degree / pooling helpers ------------------------
__global__ void fill_ones(float* p, int n) {
  const int i = blockIdx.x * blockDim.x + threadIdx.x;
  if (i < n) p[i] = 1.0f;  // self-loop contributes 1 to every degree
}
__global__ void zero_f(float* p, int n) {
  const int i = blockIdx.x * blockDim.x + threadIdx.x;
  if (i < n) p[i] = 0.0f;
}
__global__ void deg_edges(const int* __restrict__ dstI, float* __restrict__ deg) {
  const int i = blockIdx.x * blockDim.x + threadIdx.x;
  if (i < NE) atomAddF(&deg[dstI[i]], 1.0f);
}
__global__ void rsqrt_inplace(float* p, int n) {
  const int i = blockIdx.x * blockDim.x + threadIdx.x;
  if (i < n) p[i] = rsqrtf(p[i]);  // deg >= 1 always (self-loop)
}
// wave per node; lane covers 2 of 64 feats
__global__ void pool_acc(const float* __restrict__ h2,
                         const int* __restrict__ batch,
                         float* __restrict__ sums, float* __restrict__ cnt) {
  const int tid = blockIdx.x * blockDim.x + threadIdx.x;
  const int node = tid >> 5;
  const int lane = tid & 31;
  if (node >= NN) return;
  const int g = batch[node];
  const float* hp = h2 + (size_t)node * 64 + lane * 2;
  float* sp = sums + (size_t)g * 64 + lane * 2;
  atomAddF(sp, hp[0]);
  atomAddF(sp + 1, hp[1]);
  if (lane == 0) atomAddF(&cnt[g], 1.0f);
}
__global__ void pool_mean(const float* __restrict__ sums,
                          const float* __restrict__ cnt,
                          float* __restrict__ mean) {
  const int i = blockIdx.x * blockDim.x + threadIdx.x;
  if (i < NG * 64) mean[i] = sums[i] / fmaxf(cnt[i >> 6], 1.0f);
}

// ---------------------------------------------------------------------------
extern "C" void kernel_launch(void* const* d_in, const int* in_sizes, int n_in,
                              void* d_out, int out_size, void* d_ws,
                              size_t ws_size, hipStream_t stream) {
  const float* x    = (const float*)d_in[0];
  const int*   esrc = (const int*)d_in[1];
  const int*   edst = esrc + NE;
  const int*   batch = (const int*)d_in[2];
  const float* W1 = (const float*)d_in[3];
  const float* b1 = (const float*)d_in[4];
  const float* W2 = (const float*)d_in[5];
  const float* b2 = (const float*)d_in[6];
  const float* Wl = (const float*)d_in[7];
  const float* bl = (const float*)d_in[8];
  float* out = (float*)d_out;

  // workspace carve-up (floats): ~52 MB total, fits L2
  float* ws     = (float*)d_ws;
  float* dinv   = ws;                                  // NN (padded to 50048)
  float* hbuf   = ws + 50048;                          // NN*128: hn1 -> h1 -> h2
  float* aggbuf = hbuf + (size_t)NN * 128;             // NN*128: agg1; reused L2
  float* agg2   = aggbuf;                              // NN*64
  float* hn2    = aggbuf + (size_t)NN * 64;            // NN*64
  float* sums   = aggbuf + (size_t)NN * 128;           // NG*64
  float* cnt    = sums + NG * 64;                      // NG
  float* mean   = cnt + NG;                            // NG*64

  // degrees (shared by both layers): deg = 1 + in-degree; dinv = rsqrt(deg)
  fill_ones<<<(NN + 255) / 256, 256, 0, stream>>>(dinv, NN);
  deg_edges<<<(NE + 255) / 256, 256, 0, stream>>>(edst, dinv);
  rsqrt_inplace<<<(NN + 255) / 256, 256, 0, stream>>>(dinv, NN);

  // ---- layer 1: hn1 = (x@W1)*dinv ; agg1 = hn1 + scatter ; h1 = relu(...)
  gemm_scaled_dual<128, 128>
      <<<dim3(NN / 16, 128 / 16), 32, 0, stream>>>(x, W1, dinv, hbuf, aggbuf);
  scatter_edges<4><<<(NE * 32) / 256, 256, 0, stream>>>(esrc, edst, hbuf, aggbuf);
  finalize_nodes<128, true>
      <<<(NN * 128) / 256, 256, 0, stream>>>(aggbuf, dinv, b1, hbuf);

  // ---- layer 2: same with 64 feats, no relu
  gemm_scaled_dual<128, 64>
      <<<dim3(NN / 16, 64 / 16), 32, 0, stream>>>(hbuf, W2, dinv, hn2, agg2);
  scatter_edges<2><<<(NE * 32) / 256, 256, 0, stream>>>(esrc, edst, hn2, agg2);
  finalize_nodes<64, false>
      <<<(NN * 64) / 256, 256, 0, stream>>>(agg2, dinv, b2, hbuf);

  // ---- global mean pool over batch
  zero_f<<<(NG * 64 + NG + 255) / 256, 256, 0, stream>>>(sums, NG * 64 + NG);
  pool_acc<<<(NN * 32 + 255) / 256, 256, 0, stream>>>(hbuf, batch, sums, cnt);
  pool_mean<<<(NG * 64) / 256, 256, 0, stream>>>(sums, cnt, mean);

  // ---- classifier head: out = mean @ Wl + bl
  gemm_bias<64, 64><<<dim3(NG / 16, 64 / 16), 32, 0, stream>>>(mean, Wl, bl, out);
}